// SimpleMovingAverage_22144851378808
// MI455X (gfx1250) — compile-verified
//
#include <hip/hip_runtime.h>
#include <hip/hip_bf16.h>

typedef __attribute__((ext_vector_type(2))) float v2f;
typedef __attribute__((ext_vector_type(8))) float v8f;

#define Q       12
#define T_OUT   336
#define NCOL    2586      // 862*3 contiguous columns per (batch, time) row
#define NTILES  162       // ceil(2586/16)
#define MTILES  21        // 336/16
#define BATCH   64

// ---------------------------------------------------------------------------
// Kernel 1: build the 336x12 coefficient matrix of the linear recurrence.
// Thread j runs the moving-average recurrence with initial window e_j; by
// linearity the emitted means are exactly C[t][j]. Mean is order-invariant,
// so each step just overwrites slot (t % 12) with the new mean (no shifts);
// the 12-phase inner loop is fully unrolled so w[] lives in registers.
// ---------------------------------------------------------------------------
__global__ __launch_bounds__(32) void sma_coef_kernel(float* __restrict__ Ct) {
    const int j = threadIdx.x;
    if (j >= Q) return;
    float w[Q];
#pragma unroll
    for (int i = 0; i < Q; ++i) w[i] = (i == j) ? 1.0f : 0.0f;
    const float inv = 1.0f / 12.0f;
    for (int outer = 0; outer < T_OUT / Q; ++outer) {
#pragma unroll
        for (int k = 0; k < Q; ++k) {
            float s = 0.0f;
#pragma unroll
            for (int i = 0; i < Q; ++i) s += w[i];
            const float m = s * inv;
            Ct[(outer * Q + k) * Q + j] = m;
            w[k] = m;
        }
    }
}

// ---------------------------------------------------------------------------
// Kernel 2: per batch b, Out_b(336x2586) = C(336x12) @ H_b(12x2586) via
// V_WMMA_F32_16X16X4_F32, K=12 as three k-steps of 4.
// One wave per (batch, n-tile); loop over 21 m-tiles.
//
// f32 16x4 A operand layout (ISA 7.12.2): VGPR0 = K=k0 (lanes 0-15) / K=k0+2
// (lanes 16-31); VGPR1 = K=k0+1 / k0+3. B (4x16) mirrors it with N in lanes.
// C/D: VGPR r = row r (lanes 0-15) / row r+8 (lanes 16-31), N = lane&15.
// ---------------------------------------------------------------------------
__global__ __launch_bounds__(256) void sma_wmma_kernel(
    const float* __restrict__ hist,   // (64, 336, 2586)
    const float* __restrict__ Ct,     // (336, 12)
    float* __restrict__ out)          // (64, 336, 2586)
{
    const int lane  = threadIdx.x & 31;
    const int wv    = threadIdx.x >> 5;
    const int gw    = blockIdx.x * 8 + wv;          // 64*162 waves, exact fit
    const int b     = gw / NTILES;
    const int ntile = gw % NTILES;

    const int  half = lane >> 4;                    // 0: lanes 0-15, 1: 16-31
    const int  l16  = lane & 15;
    const int  col  = ntile * 16 + l16;
    const bool cok  = (col < NCOL);

    // B operands: H[time = (336-12) + k][col], shared across all m-tiles.
    const float* hb = hist + (size_t)(b * T_OUT + (T_OUT - Q)) * NCOL;
    v2f Bm[3];
#pragma unroll
    for (int ks = 0; ks < 3; ++ks) {
        const int k0 = ks * 4 + 2 * half;
        Bm[ks].x = cok ? hb[(size_t)k0 * NCOL + col]       : 0.0f;
        Bm[ks].y = cok ? hb[(size_t)(k0 + 1) * NCOL + col] : 0.0f;
    }

    float* ob = out + (size_t)b * T_OUT * NCOL;

    for (int mt = 0; mt < MTILES; ++mt) {
        const int row = mt * 16 + l16;              // M index for A loads
        v2f Am[3];
#pragma unroll
        for (int ks = 0; ks < 3; ++ks) {
            const int k0 = ks * 4 + 2 * half;
            Am[ks].x = Ct[row * Q + k0];
            Am[ks].y = Ct[row * Q + k0 + 1];
        }

        v8f acc = {};
        acc = __builtin_amdgcn_wmma_f32_16x16x4_f32(false, Am[0], false, Bm[0],
                                                    (short)0, acc, false, false);
        acc = __builtin_amdgcn_wmma_f32_16x16x4_f32(false, Am[1], false, Bm[1],
                                                    (short)0, acc, false, false);
        acc = __builtin_amdgcn_wmma_f32_16x16x4_f32(false, Am[2], false, Bm[2],
                                                    (short)0, acc, false, false);

        if (cok) {
#pragma unroll
            for (int r = 0; r < 8; ++r) {
                const int t = mt * 16 + r + 8 * half;
                __builtin_nontemporal_store(acc[r], &ob[(size_t)t * NCOL + col]);
            }
        }
    }
}

extern "C" void kernel_launch(void* const* d_in, const int* in_sizes, int n_in,
                              void* d_out, int out_size, void* d_ws, size_t ws_size,
                              hipStream_t stream) {
    const float* hist = (const float*)d_in[0];
    float*       out  = (float*)d_out;
    float*       Ct   = (float*)d_ws;   // 336*12 floats = 16128 bytes

    sma_coef_kernel<<<1, 32, 0, stream>>>(Ct);

    const int waves  = BATCH * NTILES;       // 10368
    const int blocks = waves / 8;            // 1296, exact
    sma_wmma_kernel<<<blocks, 256, 0, stream>>>(hist, Ct, out);
}